// SpatialTemporalSamplingv2_90288802496926
// MI455X (gfx1250) — compile-verified
//
#include <hip/hip_runtime.h>

// ---------------------------------------------------------------------------
// SpatialTemporalSampling v2 for MI455X (gfx1250, wave32).
// Deformable convs -> bilinear sample->LDS + V_WMMA_F32_16X16X4_F32 GEMMs
// (fp32 end-to-end, matching the fp32 reference).
//  * sample params precomputed per (img,k,px), reused over all channels
//  * 64x16 val chunk in LDS shared by all 8 waves of a block
//  * A/B preloaded into registers per chunk -> 16 back-to-back WMMAs
//  * 1x1-conv B tile staged with global_load_async_to_lds_b128 (ASYNCcnt)
// ---------------------------------------------------------------------------

#define HH    64
#define WWD   64
#define HW    4096          // 64*64
#define NIMG  12            // B*F = 4*3
#define CFEAT 256           // 2*C feature channels
#define K9    9
#define KOFF  18

typedef float v2f __attribute__((ext_vector_type(2)));
typedef float v8f __attribute__((ext_vector_type(8)));

struct SP { int y0; int x0; float wy; float wx; };   // 16 bytes

// ---------------------------------------------------------------------------
// feat[img, ch, px]:  ch<128 -> frame 0, ch>=128 -> frame f   (img = b*3+f)
__global__ void build_feat_k(const float* __restrict__ in, float* __restrict__ feat) {
    int i   = blockIdx.x * 256 + threadIdx.x;      // over NIMG*256*HW
    int px  = i & (HW - 1);
    int t   = i >> 12;                             // img*256 + ch
    int ch  = t & 255;
    int img = t >> 8;
    int b   = img / 3, f = img % 3;
    int c   = ch & 127;
    int fr  = (ch < 128) ? 0 : f;
    feat[i] = in[((b * 128 + c) * 3 + fr) * HW + px];
}

// f[img, c, px] = inputs[b, c, f, px]   (img = b*3+f, 128 channels)
__global__ void build_f_k(const float* __restrict__ in, float* __restrict__ f) {
    int i   = blockIdx.x * 256 + threadIdx.x;      // over NIMG*128*HW
    int px  = i & (HW - 1);
    int t   = i >> 12;
    int c   = t & 127;
    int img = t >> 7;
    int b   = img / 3, fr = img % 3;
    f[i] = in[((b * 128 + c) * 3 + fr) * HW + px];
}

// ---------------------------------------------------------------------------
// Direct 3x3 conv (pad 1) -> 18 offset channels, + bias, ReLU.
__global__ void conv3_off_k(const float* __restrict__ in, const float* __restrict__ w,
                            const float* __restrict__ bias, float* __restrict__ out,
                            int Cin) {
    int px  = blockIdx.x * 256 + threadIdx.x;
    int oc  = blockIdx.y;
    int img = blockIdx.z;
    int y = px >> 6, x = px & 63;
    const float* ib = in + (size_t)img * Cin * HW;
    const float* wb = w  + (size_t)oc  * Cin * 9;
    float acc = bias[oc];
    for (int c = 0; c < Cin; ++c) {
        const float* ic = ib + c * HW;
        const float* wc = wb + c * 9;
        // batch the 9 taps: load first, combine after (one wait, not nine)
        float t[9]; float wv[9];
        #pragma unroll
        for (int ky = 0; ky < 3; ++ky) {
            #pragma unroll
            for (int kx = 0; kx < 3; ++kx) {
                int yy = y + ky - 1, xx = x + kx - 1;
                bool ok = ((unsigned)yy < HH) && ((unsigned)xx < WWD);
                int yyc = min(max(yy, 0), HH - 1);
                int xxc = min(max(xx, 0), WWD - 1);
                t[ky * 3 + kx]  = ic[yyc * WWD + xxc];
                wv[ky * 3 + kx] = ok ? wc[ky * 3 + kx] : 0.f;
            }
        }
        #pragma unroll
        for (int j = 0; j < 9; ++j) acc += wv[j] * t[j];
    }
    out[((size_t)img * KOFF + oc) * HW + px] = fmaxf(acc, 0.f);
}

// ---------------------------------------------------------------------------
// Bilinear sample parameters: one per (img, k, px), reused across channels.
__global__ void sparams_k(const float* __restrict__ off, SP* __restrict__ sp) {
    int i   = blockIdx.x * 256 + threadIdx.x;      // over NIMG*9*HW
    int px  = i & (HW - 1);
    int t   = i >> 12;
    int k   = t % 9;
    int img = t / 9;
    float dy = off[((size_t)img * KOFF + 2 * k    ) * HW + px];
    float dx = off[((size_t)img * KOFF + 2 * k + 1) * HW + px];
    int y = px >> 6, x = px & 63;
    float ys = (float)y - 1.f + (float)(k / 3) + dy;
    float xs = (float)x - 1.f + (float)(k % 3) + dx;
    float y0f = floorf(ys), x0f = floorf(xs);
    SP s; s.y0 = (int)y0f; s.x0 = (int)x0f; s.wy = ys - y0f; s.wx = xs - x0f;
    sp[i] = s;
}

// ---------------------------------------------------------------------------
// Fused deformable-conv GEMM: block = 8 waves, 16-pixel tile, 128 out chans.
__global__ void __launch_bounds__(256)
deform_wmma_k(const float* __restrict__ in, const SP* __restrict__ sp,
              const float* __restrict__ w,  const float* __restrict__ bias,
              float* __restrict__ out, int Cin, int OC, int relu) {
    __shared__ SP    s_sp[K9 * 16];
    __shared__ float s_B[64 * 16];

    const int tid     = threadIdx.x;
    const int img     = blockIdx.z;
    const int px_base = blockIdx.x * 16;
    const int o_tile  = blockIdx.y * 128 + (tid >> 5) * 16;
    const int lane    = tid & 31;
    const int lo      = lane & 15;        // N column / A row within tile
    const int hi      = lane >> 4;        // K half selector (layout tables)

    if (tid < K9 * 16) {
        int k = tid >> 4, p = tid & 15;
        s_sp[tid] = sp[((size_t)img * K9 + k) * HW + px_base + p];
    }

    const int CK = Cin * 9;               // 2304 or 1152, both %64 == 0
    const float* wrow  = w  + (size_t)(o_tile + lo) * CK;
    const float* ibase = in + (size_t)img * Cin * HW;

    v8f acc = {};
    for (int ckb = 0; ckb < CK; ckb += 64) {
        __syncthreads();                  // protect s_B from previous round
        // ---- phase 1: tap pointers + weights for this thread's 4 entries --
        const float* tp[16];
        float        tw[16];
        #pragma unroll
        for (int t = 0; t < 4; ++t) {
            int e   = tid + t * 256;
            int ckl = e >> 4, p = e & 15;
            int ck  = ckb + ckl;
            int c   = ck / 9;
            int k   = ck - c * 9;
            SP s = s_sp[k * 16 + p];
            const float* ch = ibase + (size_t)c * HW;
            int y0 = s.y0, x0 = s.x0, y1 = y0 + 1, x1 = x0 + 1;
            float wy = s.wy, wx = s.wx;
            float w00 = (1.f - wy) * (1.f - wx);
            float w01 = (1.f - wy) * wx;
            float w10 = wy * (1.f - wx);
            float w11 = wy * wx;
            if ((unsigned)y0 >= HH)  { w00 = 0.f; w01 = 0.f; }
            if ((unsigned)y1 >= HH)  { w10 = 0.f; w11 = 0.f; }
            if ((unsigned)x0 >= WWD) { w00 = 0.f; w10 = 0.f; }
            if ((unsigned)x1 >= WWD) { w01 = 0.f; w11 = 0.f; }
            int y0c = min(max(y0, 0), HH - 1),  y1c = min(max(y1, 0), HH - 1);
            int x0c = min(max(x0, 0), WWD - 1), x1c = min(max(x1, 0), WWD - 1);
            tp[t * 4 + 0] = ch + y0c * WWD + x0c;  tw[t * 4 + 0] = w00;
            tp[t * 4 + 1] = ch + y0c * WWD + x1c;  tw[t * 4 + 1] = w01;
            tp[t * 4 + 2] = ch + y1c * WWD + x0c;  tw[t * 4 + 2] = w10;
            tp[t * 4 + 3] = ch + y1c * WWD + x1c;  tw[t * 4 + 3] = w11;
        }
        // ---- phase 2: 16 loads in flight, one wait ------------------------
        float tap[16];
        #pragma unroll
        for (int j = 0; j < 16; ++j) tap[j] = *tp[j];
        // ---- phase 3: combine + store to LDS ------------------------------
        #pragma unroll
        for (int t = 0; t < 4; ++t) {
            int e = tid + t * 256;
            float v = tw[t*4+0]*tap[t*4+0] + tw[t*4+1]*tap[t*4+1]
                    + tw[t*4+2]*tap[t*4+2] + tw[t*4+3]*tap[t*4+3];
            s_B[(e >> 4) * 16 + (e & 15)] = v;
        }
        __syncthreads();
        // ---- WMMA phase: preload A+B for the whole chunk, then stream -----
        v2f a[16], bb[16];
        const float* wp = wrow + ckb + 2 * hi;
        #pragma unroll
        for (int j = 0; j < 16; ++j) a[j] = *(const v2f*)(wp + 4 * j);
        #pragma unroll
        for (int j = 0; j < 16; ++j) {
            int kk = 4 * j + 2 * hi;
            bb[j].x = s_B[kk * 16 + lo];
            bb[j].y = s_B[(kk + 1) * 16 + lo];
        }
        #pragma unroll
        for (int j = 0; j < 16; ++j)
            acc = __builtin_amdgcn_wmma_f32_16x16x4_f32(
                      false, a[j], false, bb[j], (short)0, acc, false, false);
    }
    // --- D store: VGPR r -> M = r + 8*hi, N = lo (ISA C/D layout) ---
    #pragma unroll
    for (int r = 0; r < 8; ++r) {
        int oc = o_tile + r + 8 * hi;
        float v = acc[r];
        if (bias) v += bias[oc];
        if (relu) v = fmaxf(v, 0.f);
        out[((size_t)img * OC + oc) * HW + px_base + lo] = v;
    }
}

// ---------------------------------------------------------------------------
// BatchNorm: per-channel mean/var over (N=12, H, W) then affine apply.
__global__ void bn_reduce_k(const float* __restrict__ x, float* __restrict__ stats) {
    int ch = blockIdx.x, tid = threadIdx.x;
    float s = 0.f, s2 = 0.f;
    for (int img = 0; img < NIMG; ++img) {
        const float* p = x + ((size_t)img * CFEAT + ch) * HW;
        for (int i = tid; i < HW; i += 256) {
            float v = p[i]; s += v; s2 += v * v;
        }
    }
    __shared__ float r1[256], r2[256];
    r1[tid] = s; r2[tid] = s2; __syncthreads();
    for (int o = 128; o > 0; o >>= 1) {
        if (tid < o) { r1[tid] += r1[tid + o]; r2[tid] += r2[tid + o]; }
        __syncthreads();
    }
    if (tid == 0) {
        float n = (float)(NIMG * HW);
        float m = r1[0] / n;
        stats[ch]         = m;
        stats[CFEAT + ch] = r2[0] / n - m * m;   // biased var (ddof=0)
    }
}

__global__ void bn_apply_k(float* __restrict__ x, const float* __restrict__ stats,
                           const float* __restrict__ g, const float* __restrict__ b) {
    int i  = blockIdx.x * 256 + threadIdx.x;     // over NIMG*256*HW
    int ch = (i >> 12) & 255;
    float inv = rsqrtf(stats[CFEAT + ch] + 1e-5f);
    x[i] = (x[i] - stats[ch]) * inv * g[ch] + b[ch];
}

// ---------------------------------------------------------------------------
// Final 1x1 conv: out[b,o,px] = sum_i mw[o,i]*x[b,i,px] + mb[o]
// GEMM M=128, K=384, N=4096 per batch. B tile staged with async-to-LDS.
__global__ void __launch_bounds__(256)
pw_wmma_k(const float* __restrict__ x, const float* __restrict__ w,
          const float* __restrict__ bias, float* __restrict__ out) {
    __shared__ float s_B[64 * 16];
    const int tid     = threadIdx.x;
    const int bidx    = blockIdx.z;
    const int px_base = blockIdx.x * 16;
    const int o_tile  = (tid >> 5) * 16;         // 8 waves x 16 = 128 = OC
    const int lane    = tid & 31;
    const int lo      = lane & 15;
    const int hi      = lane >> 4;
    const float* wrow = w + (size_t)(o_tile + lo) * 384;
    const float* xb   = x + (size_t)bidx * 384 * HW;

    // per-thread async copy slot: row = tid>>2 (64 rows), group = tid&3 (4x16B)
    const int row = tid >> 2, grp = tid & 3;
    const unsigned lds_off =
        (unsigned)(uintptr_t)(&s_B[row * 16 + grp * 4]);   // flat low 32 = LDS offset

    v8f acc = {};
    for (int kb = 0; kb < 384; kb += 64) {
        __syncthreads();
        // --- async DMA of the 64x16 B tile into LDS (16B per thread) -------
        const float* gp = xb + (size_t)(kb + row) * HW + px_base + grp * 4;
        asm volatile("global_load_async_to_lds_b128 %0, %1, off"
                     :: "v"(lds_off), "v"(gp) : "memory");
        asm volatile("s_wait_asynccnt 0x0" ::: "memory");
        __syncthreads();
        // --- preload A+B chunk, then 16 back-to-back WMMAs -----------------
        v2f a[16], bb[16];
        const float* wp = wrow + kb + 2 * hi;
        #pragma unroll
        for (int j = 0; j < 16; ++j) a[j] = *(const v2f*)(wp + 4 * j);
        #pragma unroll
        for (int j = 0; j < 16; ++j) {
            int kk = 4 * j + 2 * hi;
            bb[j].x = s_B[kk * 16 + lo];
            bb[j].y = s_B[(kk + 1) * 16 + lo];
        }
        #pragma unroll
        for (int j = 0; j < 16; ++j)
            acc = __builtin_amdgcn_wmma_f32_16x16x4_f32(
                      false, a[j], false, bb[j], (short)0, acc, false, false);
    }
    #pragma unroll
    for (int r = 0; r < 8; ++r) {
        int oc = o_tile + r + 8 * hi;
        out[((size_t)bidx * 128 + oc) * HW + px_base + lo] = acc[r] + bias[oc];
    }
}

// ---------------------------------------------------------------------------
extern "C" void kernel_launch(void* const* d_in, const int* in_sizes, int n_in,
                              void* d_out, int out_size, void* d_ws, size_t ws_size,
                              hipStream_t stream) {
    const float* inputs = (const float*)d_in[0];
    const float* ow0 = (const float*)d_in[1];
    const float* ob0 = (const float*)d_in[2];
    const float* dw0 = (const float*)d_in[3];
    const float* ow1 = (const float*)d_in[4];
    const float* ob1 = (const float*)d_in[5];
    const float* dw1 = (const float*)d_in[6];
    const float* owf = (const float*)d_in[7];
    const float* obf = (const float*)d_in[8];
    const float* dwf = (const float*)d_in[9];
    const float* dbf = (const float*)d_in[10];
    const float* g0  = (const float*)d_in[11];
    const float* b0  = (const float*)d_in[12];
    const float* g1  = (const float*)d_in[13];
    const float* b1  = (const float*)d_in[14];
    const float* mw  = (const float*)d_in[15];
    const float* mb  = (const float*)d_in[16];

    // workspace carve-up (≈136 MB total)
    char* ws = (char*)d_ws;
    float* featA = (float*)ws;  ws += (size_t)NIMG * 256 * HW * 4;   // 50.3 MB
    float* featB = (float*)ws;  ws += (size_t)NIMG * 256 * HW * 4;   // 50.3 MB
    float* off   = (float*)ws;  ws += (size_t)NIMG * KOFF * HW * 4;  //  3.5 MB
    SP*    sp    = (SP*)ws;     ws += (size_t)NIMG * K9 * HW * 16;   //  7.1 MB
    float* fout  = (float*)ws;  ws += (size_t)NIMG * 128 * HW * 4;   // 25.2 MB
    float* stats = (float*)ws;                                       //  2 KB

    const dim3 blk(256);
    const dim3 gOff(HW / 256, KOFF, NIMG);
    const dim3 gSp((NIMG * K9 * HW) / 256);
    const dim3 gDef(HW / 16, 2, NIMG);    // OC=256 -> 2 oc-blocks of 128
    const dim3 gDefF(HW / 16, 1, NIMG);   // OC=128
    const dim3 gBnR(256);
    const int  nFeat = (NIMG * 256 * HW) / 256;

    build_feat_k<<<nFeat, blk, 0, stream>>>(inputs, featA);

    // ---- layer 0: featA -> featB ----
    conv3_off_k<<<gOff, blk, 0, stream>>>(featA, ow0, ob0, off, CFEAT);
    sparams_k<<<gSp, blk, 0, stream>>>(off, sp);
    deform_wmma_k<<<gDef, blk, 0, stream>>>(featA, sp, dw0, nullptr, featB,
                                            CFEAT, 256, 1);
    bn_reduce_k<<<gBnR, blk, 0, stream>>>(featB, stats);
    bn_apply_k<<<nFeat, blk, 0, stream>>>(featB, stats, g0, b0);

    // ---- layer 1: featB -> featA ----
    conv3_off_k<<<gOff, blk, 0, stream>>>(featB, ow1, ob1, off, CFEAT);
    sparams_k<<<gSp, blk, 0, stream>>>(off, sp);
    deform_wmma_k<<<gDef, blk, 0, stream>>>(featB, sp, dw1, nullptr, featA,
                                            CFEAT, 256, 1);
    bn_reduce_k<<<gBnR, blk, 0, stream>>>(featA, stats);
    bn_apply_k<<<nFeat, blk, 0, stream>>>(featA, stats, g1, b1);

    // ---- final: offsets from featA, deform on raw frames, 1x1 fuse ----
    conv3_off_k<<<gOff, blk, 0, stream>>>(featA, owf, obf, off, CFEAT);
    sparams_k<<<gSp, blk, 0, stream>>>(off, sp);
    build_f_k<<<(NIMG * 128 * HW) / 256, blk, 0, stream>>>(inputs, featB);
    deform_wmma_k<<<gDefF, blk, 0, stream>>>(featB, sp, dwf, dbf, fout,
                                             128, 128, 0);
    // fout viewed as (4, 384, HW) is exactly the reshape(B, F*OC, H, W)
    pw_wmma_k<<<dim3(HW / 16, 1, 4), blk, 0, stream>>>(fout, mw, mb,
                                                       (float*)d_out);
}